// DiffTransformerLayer_65146063946370
// MI455X (gfx1250) — compile-verified
//
#include <hip/hip_runtime.h>

// Problem constants (from the reference)
#define BB 4
#define SS 1024
#define DD 1024
#define HH 16
#define HDD 32          // head dim per component
#define FF_ 4096
static constexpr float LAMBDA_INIT_F = 0.78360575437545776f; // 0.8 - 0.6*exp(-3.6)
static constexpr float EPS_F = 1e-5f;

typedef __attribute__((ext_vector_type(2))) float v2f;
typedef __attribute__((ext_vector_type(8))) float v8f;

__device__ __forceinline__ v8f wmma_f32(v2f a, v2f b, v8f c) {
    // D = A(16x4 f32) * B(4x16 f32) + C(16x16 f32)
    return __builtin_amdgcn_wmma_f32_16x16x4_f32(false, a, false, b, (short)0, c, false, false);
}

// ---- CDNA5 async global->LDS copy support (gated so the file always builds) ----
#if defined(__has_builtin)
#if __has_builtin(__builtin_amdgcn_global_load_async_to_lds_b128)
#define USE_ASYNC_LDS 1
#endif
#endif
#ifndef USE_ASYNC_LDS
#define USE_ASYNC_LDS 0
#endif

#if USE_ASYNC_LDS
typedef int v4i_vs __attribute__((vector_size(4 * sizeof(int))));
typedef __attribute__((address_space(1))) v4i_vs* gv4_ptr;  // global int4*
typedef __attribute__((address_space(3))) v4i_vs* lv4_ptr;  // LDS int4*
__device__ __forceinline__ void async_copy16(const float* g, float* l) {
    __builtin_amdgcn_global_load_async_to_lds_b128((gv4_ptr)g, (lv4_ptr)l, 0, 0);
}
__device__ __forceinline__ void wait_async0() {
#if __has_builtin(__builtin_amdgcn_s_wait_asynccnt)
    __builtin_amdgcn_s_wait_asynccnt(0);
#else
    asm volatile("s_wait_asynccnt 0" ::: "memory");
#endif
}
#endif

// ---------------------------------------------------------------------------
// Generic C = A @ W^T (+bias)(+relu)(+residual).  A: MxK, W: NxK, C: MxN.
// Block: 256 threads = 8 waves. Tile 128x128, K-stage 16, double-buffered LDS
// filled with GLOBAL_LOAD_ASYNC_TO_LDS_B128 (ASYNCcnt) when available.
// Wave (wm 0..3, wn 0..1) computes rows [wm*32,+32), cols [wn*64,+64):
// 2x4 f32 WMMA accumulators per wave -> 32 v_wmma per K-stage per wave.
// ---------------------------------------------------------------------------
template <bool BIAS, bool RELU, bool RES>
__global__ __launch_bounds__(256) void gemm_nt(
    const float* __restrict__ A, const float* __restrict__ W,
    const float* __restrict__ bias, const float* __restrict__ res,
    float* __restrict__ C, int M, int N, int K)
{
    __shared__ float As[2][128][16];
    __shared__ float Bs[2][128][16];

    const int tid  = threadIdx.x;
    const int lane = tid & 31;
    const int wid  = tid >> 5;
    const int m    = lane & 15;   // A-frag row / B-frag col
    const int hi   = lane >> 4;   // lane half selects K pair {0,1} vs {2,3}
    const int wm   = wid >> 1;    // 0..3
    const int wn   = wid & 1;     // 0..1

    const int n0 = blockIdx.x * 128;
    const int m0 = blockIdx.y * 128;

    const int lrow = tid >> 1;        // 0..127
    const int lcol = (tid & 1) * 8;   // 0 or 8

    const float* Ag = A + (size_t)(m0 + lrow) * K + lcol;
    const float* Wg = W + (size_t)(n0 + lrow) * K + lcol;

    v8f acc[2][4];
#pragma unroll
    for (int mi = 0; mi < 2; ++mi)
#pragma unroll
        for (int ni = 0; ni < 4; ++ni) { v8f z = {}; acc[mi][ni] = z; }

    auto stage = [&](int buf, int k0) {
#if USE_ASYNC_LDS
        async_copy16(Ag + k0,     &As[buf][lrow][lcol]);
        async_copy16(Ag + k0 + 4, &As[buf][lrow][lcol + 4]);
        async_copy16(Wg + k0,     &Bs[buf][lrow][lcol]);
        async_copy16(Wg + k0 + 4, &Bs[buf][lrow][lcol + 4]);
#else
        float4 a0 = *(const float4*)(Ag + k0);
        float4 a1 = *(const float4*)(Ag + k0 + 4);
        float4 b0 = *(const float4*)(Wg + k0);
        float4 b1 = *(const float4*)(Wg + k0 + 4);
        *(float4*)&As[buf][lrow][lcol]     = a0;
        *(float4*)&As[buf][lrow][lcol + 4] = a1;
        *(float4*)&Bs[buf][lrow][lcol]     = b0;
        *(float4*)&Bs[buf][lrow][lcol + 4] = b1;
#endif
    };

    stage(0, 0);
#if USE_ASYNC_LDS
    wait_async0();
#endif
    __syncthreads();

    int buf = 0;
    for (int k0 = 0; k0 < K; k0 += 16) {
        if (k0 + 16 < K) stage(buf ^ 1, k0 + 16);
#pragma unroll
        for (int kk = 0; kk < 4; ++kk) {
            v2f af0 = *(const v2f*)&As[buf][wm * 32 + m][kk * 4 + 2 * hi];
            v2f af1 = *(const v2f*)&As[buf][wm * 32 + 16 + m][kk * 4 + 2 * hi];
            v2f bf[4];
#pragma unroll
            for (int ni = 0; ni < 4; ++ni)
                bf[ni] = *(const v2f*)&Bs[buf][wn * 64 + ni * 16 + m][kk * 4 + 2 * hi];
#pragma unroll
            for (int ni = 0; ni < 4; ++ni) {
                acc[0][ni] = wmma_f32(af0, bf[ni], acc[0][ni]);
                acc[1][ni] = wmma_f32(af1, bf[ni], acc[1][ni]);
            }
        }
#if USE_ASYNC_LDS
        wait_async0();
#endif
        __syncthreads();
        buf ^= 1;
    }

#pragma unroll
    for (int ni = 0; ni < 4; ++ni) {
        const int gc = n0 + wn * 64 + ni * 16 + m;
        float bval = 0.0f;
        if constexpr (BIAS) bval = bias[gc];
#pragma unroll
        for (int mi = 0; mi < 2; ++mi) {
#pragma unroll
            for (int r = 0; r < 8; ++r) {
                const int gr = m0 + wm * 32 + mi * 16 + r + 8 * hi;
                float v = acc[mi][ni][r];
                if constexpr (BIAS) v += bval;
                if constexpr (RES)  v += res[(size_t)gr * N + gc];
                if constexpr (RELU) v = fmaxf(v, 0.0f);
                C[(size_t)gr * N + gc] = v;
            }
        }
    }
}

// ---------------------------------------------------------------------------
// RoPE over q and k, layout (B,S,2H,HD), pairs (2j, 2j+1), half = HD/2 = 16.
// ---------------------------------------------------------------------------
__global__ __launch_bounds__(256) void rope_kernel(float* __restrict__ q, float* __restrict__ k)
{
    const int total = BB * SS * 2 * HH * (HDD / 2); // 2,097,152
    int idx = blockIdx.x * blockDim.x + threadIdx.x;
    if (idx >= total) return;
    const int j    = idx & 15;
    const int rest = idx >> 4;              // (b*S+s)*2H + hh
    const int s    = (rest >> 5) & (SS - 1);
    const float freq = __expf(-0.57564627324851142f * (float)j); // 10000^(-j/16)
    const float ang  = (float)s * freq;
    float sn, cs;
    __sincosf(ang, &sn, &cs);
    const size_t base = (size_t)rest * HDD + 2 * j;
    float x1 = q[base], x2 = q[base + 1];
    q[base]     = x1 * cs - x2 * sn;
    q[base + 1] = x1 * sn + x2 * cs;
    x1 = k[base]; x2 = k[base + 1];
    k[base]     = x1 * cs - x2 * sn;
    k[base + 1] = x1 * sn + x2 * cs;
}

// ---------------------------------------------------------------------------
// lambda = exp(lq1.lk1) - exp(lq2.lk2) + LAMBDA_INIT   (HD = 32 = one wave)
// ---------------------------------------------------------------------------
__global__ __launch_bounds__(32) void lambda_kernel(
    const float* __restrict__ lq1, const float* __restrict__ lk1,
    const float* __restrict__ lq2, const float* __restrict__ lk2,
    float* __restrict__ out)
{
    const int t = threadIdx.x;
    float d1 = lq1[t] * lk1[t];
    float d2 = lq2[t] * lk2[t];
#pragma unroll
    for (int off = 16; off; off >>= 1) {
        d1 += __shfl_down(d1, off);
        d2 += __shfl_down(d2, off);
    }
    if (t == 0) out[0] = expf(d1) - expf(d2) + LAMBDA_INIT_F;
}

// ---------------------------------------------------------------------------
// Dual causal flash attention + combine + per-head RMS, all WMMA.
// One wave per (b, h, 16-query tile). Block = 8 waves, grid = B*H*(S/16)/8 = 512.
// ---------------------------------------------------------------------------
__global__ __launch_bounds__(256) void attn_kernel(
    const float* __restrict__ q, const float* __restrict__ k,
    const float* __restrict__ v, const float* __restrict__ lam_p,
    float* __restrict__ attn)
{
    __shared__ float Pbuf[8][16][16];

    const int wid  = threadIdx.x >> 5;
    const int lane = threadIdx.x & 31;
    const int m    = lane & 15;
    const int hi   = lane >> 4;

    const int w  = blockIdx.x * 8 + wid;
    const int qt = w & 63;        // S/16 = 64 query tiles
    const int bh = w >> 6;
    const int b  = bh >> 4;
    const int h  = bh & 15;

    const float lam   = lam_p[0];
    const float scale = 0.17677669529663687f; // 32^-0.5

    const float* qp = q + (size_t)b * SS * DD + (size_t)(2 * h) * HDD;
    const float* kp = k + (size_t)b * SS * DD + (size_t)(2 * h) * HDD;
    const float* vp = v + (size_t)b * SS * DD + (size_t)h * (2 * HDD);

    const int q0 = qt * 16;

    v2f q1a[8], q2a[8];
#pragma unroll
    for (int kk = 0; kk < 8; ++kk) {
        const float* r1 = qp + (size_t)(q0 + m) * DD + kk * 4 + 2 * hi;
        q1a[kk] = *(const v2f*)r1;
        q2a[kk] = *(const v2f*)(r1 + HDD);
    }

    v8f o1[4], o2[4];
#pragma unroll
    for (int t = 0; t < 4; ++t) { v8f z = {}; o1[t] = z; o2[t] = z; }
    float m1[8], l1[8], m2[8], l2[8];
#pragma unroll
    for (int r = 0; r < 8; ++r) { m1[r] = -3.0e38f; m2[r] = -3.0e38f; l1[r] = 0.f; l2[r] = 0.f; }

    for (int kt = 0; kt <= qt; ++kt) {
        const int kbase = kt * 16;
        v8f s1 = {}; v8f s2 = {};
#pragma unroll
        for (int kk = 0; kk < 8; ++kk) {
            const float* kr = kp + (size_t)(kbase + m) * DD + kk * 4 + 2 * hi;
            v2f kb1 = *(const v2f*)kr;
            v2f kb2 = *(const v2f*)(kr + HDD);
            s1 = wmma_f32(q1a[kk], kb1, s1);
            s2 = wmma_f32(q2a[kk], kb2, s2);
        }
        const bool diag = (kt == qt);
        float p2s[8];
#pragma unroll
        for (int r = 0; r < 8; ++r) {
            const int row = r + 8 * hi;
            float x1 = s1[r] * scale;
            float x2 = s2[r] * scale;
            if (diag && (m > row)) { x1 = -3.0e38f; x2 = -3.0e38f; }
            float r1 = x1, r2 = x2;
#pragma unroll
            for (int off = 1; off < 16; off <<= 1) {
                r1 = fmaxf(r1, __shfl_xor(r1, off));
                r2 = fmaxf(r2, __shfl_xor(r2, off));
            }
            const float mn1 = fmaxf(m1[r], r1);
            const float mn2 = fmaxf(m2[r], r2);
            const float c1 = __expf(m1[r] - mn1);
            const float c2 = __expf(m2[r] - mn2);
            const float p1 = __expf(x1 - mn1);
            const float p2 = __expf(x2 - mn2);
            float su1 = p1, su2 = p2;
#pragma unroll
            for (int off = 1; off < 16; off <<= 1) {
                su1 += __shfl_xor(su1, off);
                su2 += __shfl_xor(su2, off);
            }
            l1[r] = l1[r] * c1 + su1;
            l2[r] = l2[r] * c2 + su2;
            m1[r] = mn1; m2[r] = mn2;
#pragma unroll
            for (int t = 0; t < 4; ++t) { o1[t][r] *= c1; o2[t][r] *= c2; }
            Pbuf[wid][row][m] = p1;
            p2s[r] = p2;
        }
        // P1 @ V  (cache V B-fragments, reuse for P2 pass)
        v2f bvc[4][4];
#pragma unroll
        for (int kk = 0; kk < 4; ++kk) {
            v2f pa = *(const v2f*)&Pbuf[wid][m][kk * 4 + 2 * hi];
#pragma unroll
            for (int t = 0; t < 4; ++t) {
                const int key = kbase + kk * 4 + 2 * hi;
                const float* vb = vp + (size_t)key * DD + t * 16 + m;
                v2f bv; bv.x = vb[0]; bv.y = vb[DD];
                bvc[kk][t] = bv;
                o1[t] = wmma_f32(pa, bv, o1[t]);
            }
        }
#pragma unroll
        for (int r = 0; r < 8; ++r) Pbuf[wid][r + 8 * hi][m] = p2s[r];
#pragma unroll
        for (int kk = 0; kk < 4; ++kk) {
            v2f pa = *(const v2f*)&Pbuf[wid][m][kk * 4 + 2 * hi];
#pragma unroll
            for (int t = 0; t < 4; ++t) o2[t] = wmma_f32(pa, bvc[kk][t], o2[t]);
        }
    }

#pragma unroll
    for (int r = 0; r < 8; ++r) {
        const float inv1 = 1.0f / l1[r];
        const float inv2 = 1.0f / l2[r];
        float cv[4];
        float ssum = 0.0f;
#pragma unroll
        for (int t = 0; t < 4; ++t) {
            const float a = o1[t][r] * inv1 - lam * (o2[t][r] * inv2);
            cv[t] = a;
            ssum += a * a;
        }
#pragma unroll
        for (int off = 1; off < 16; off <<= 1) ssum += __shfl_xor(ssum, off);
        const float rn = rsqrtf(ssum * (1.0f / 64.0f) + EPS_F) * (1.0f - LAMBDA_INIT_F);
        const int row = q0 + r + 8 * hi;
#pragma unroll
        for (int t = 0; t < 4; ++t)
            attn[((size_t)b * SS + row) * DD + h * 64 + t * 16 + m] = cv[t] * rn;
    }
}

// ---------------------------------------------------------------------------
// Final RMS-norm over D=1024 with learned weight. One wave per row.
// ---------------------------------------------------------------------------
__global__ __launch_bounds__(256) void rmsnorm_kernel(
    const float* __restrict__ y, const float* __restrict__ wgt, float* __restrict__ out)
{
    const int wid  = threadIdx.x >> 5;
    const int lane = threadIdx.x & 31;
    const int row  = blockIdx.x * 8 + wid;
    const float* yr = y + (size_t)row * DD;
    float vals[32];
    float ssum = 0.0f;
#pragma unroll
    for (int i = 0; i < 32; ++i) {
        const float vv = yr[lane + i * 32];
        vals[i] = vv;
        ssum += vv * vv;
    }
#pragma unroll
    for (int off = 1; off < 32; off <<= 1) ssum += __shfl_xor(ssum, off);
    const float rn = rsqrtf(ssum * (1.0f / 1024.0f) + EPS_F);
    float* orow = out + (size_t)row * DD;
#pragma unroll
    for (int i = 0; i < 32; ++i) orow[lane + i * 32] = vals[i] * rn * wgt[lane + i * 32];
}

// ---------------------------------------------------------------------------
extern "C" void kernel_launch(void* const* d_in, const int* in_sizes, int n_in,
                              void* d_out, int out_size, void* d_ws, size_t ws_size,
                              hipStream_t stream)
{
    const float* x   = (const float*)d_in[0];
    const float* Wq  = (const float*)d_in[1];
    const float* Wk  = (const float*)d_in[2];
    const float* Wv  = (const float*)d_in[3];
    const float* Wo  = (const float*)d_in[4];
    const float* lq1 = (const float*)d_in[5];
    const float* lk1 = (const float*)d_in[6];
    const float* lq2 = (const float*)d_in[7];
    const float* lk2 = (const float*)d_in[8];
    const float* W1  = (const float*)d_in[9];
    const float* b1  = (const float*)d_in[10];
    const float* W2  = (const float*)d_in[11];
    const float* b2  = (const float*)d_in[12];
    const float* rmsw = (const float*)d_in[13];

    float* ws = (float*)d_ws;
    const size_t T4 = (size_t)BB * SS * DD; // 4 Mi floats = 16 MB

    float* qbuf     = ws;            // [0, T4)        later reused: attn_out, y
    float* kbuf     = ws + T4;       // [T4, 2T4)
    float* vbuf     = ws + 2 * T4;   // [2T4, 3T4)
    float* attnbuf  = ws + 3 * T4;   // [3T4, 4T4)
    float* h1buf    = ws + T4;       // [T4, 5T4)  (k,v,attn dead by then)
    float* attn_out = ws;            // reuses q (dead after attention)
    float* ybuf     = ws;            // in-place residual epilogue of W2 GEMM
    float* lamb     = ws + 5 * T4;   // single scalar

    const int M = BB * SS; // 4096
    dim3 blk(256);

    // QKV projections
    gemm_nt<false, false, false><<<dim3(DD / 128, M / 128), blk, 0, stream>>>(x, Wq, nullptr, nullptr, qbuf, M, DD, DD);
    gemm_nt<false, false, false><<<dim3(DD / 128, M / 128), blk, 0, stream>>>(x, Wk, nullptr, nullptr, kbuf, M, DD, DD);
    gemm_nt<false, false, false><<<dim3(DD / 128, M / 128), blk, 0, stream>>>(x, Wv, nullptr, nullptr, vbuf, M, DD, DD);

    // RoPE on q,k
    const int rope_total = BB * SS * 2 * HH * (HDD / 2);
    rope_kernel<<<(rope_total + 255) / 256, blk, 0, stream>>>(qbuf, kbuf);

    // lambda scalar
    lambda_kernel<<<1, 32, 0, stream>>>(lq1, lk1, lq2, lk2, lamb);

    // dual flash attention + combine + head RMS
    attn_kernel<<<(BB * HH * (SS / 16)) / 8, blk, 0, stream>>>(qbuf, kbuf, vbuf, lamb, attnbuf);

    // output projection
    gemm_nt<false, false, false><<<dim3(DD / 128, M / 128), blk, 0, stream>>>(attnbuf, Wo, nullptr, nullptr, attn_out, M, DD, DD);

    // FFN: relu(attn_out @ W1^T + b1)
    gemm_nt<true, true, false><<<dim3(FF_ / 128, M / 128), blk, 0, stream>>>(attn_out, W1, b1, nullptr, h1buf, M, FF_, DD);
    // h @ W2^T + b2 + attn_out  (in-place per-element residual is safe)
    gemm_nt<true, false, true><<<dim3(DD / 128, M / 128), blk, 0, stream>>>(h1buf, W2, b2, attn_out, ybuf, M, DD, FF_);

    // final RMS-norm * weight
    rmsnorm_kernel<<<M / 8, blk, 0, stream>>>(ybuf, rmsw, (float*)d_out);
}